// StructuralOnlyGCN_90529320665198
// MI455X (gfx1250) — compile-verified
//
#include <hip/hip_runtime.h>
#include <hip/hip_bf16.h>
#include <math.h>

// ---------------------------------------------------------------------------
// StructuralOnlyGCN for MI455X (gfx1250, wave32).
// GEMMs use V_WMMA_F32_16X16X4_F32 (full fp32 precision, as in the reference)
// with a branch-free unrolled inner loop: b64 global loads for A, conflict-free
// b64 LDS loads for the repacked W tile. Edge scatter = 1 thread per
// (edge, feature); f32 global atomics coalesce per destination row.
// ---------------------------------------------------------------------------

typedef float v2f __attribute__((ext_vector_type(2)));
typedef float v8f __attribute__((ext_vector_type(8)));

#define GCN_BN_EPS 1e-5f
#define N_GRAPHS_C 512

// ---------------- utility ----------------

__global__ void k_zero_f(float* __restrict__ p, long long n) {
  long long i = (long long)blockIdx.x * blockDim.x + threadIdx.x;
  if (i < n) p[i] = 0.0f;
}

__global__ void k_deg(const int* __restrict__ dst, float* __restrict__ deg, int ne) {
  int e = blockIdx.x * blockDim.x + threadIdx.x;
  if (e < ne) atomicAdd(&deg[dst[e]], 1.0f);
}

__global__ void k_dinv(float* __restrict__ deg, int n) {
  int i = blockIdx.x * blockDim.x + threadIdx.x;
  if (i < n) deg[i] = rsqrtf(deg[i] + 1.0f);
}

// ---------------- WMMA fp32 GEMM:  C[M x NCOLS] = A[M x K] @ W[K x NCOLS] ---
// One wave computes one 16x16 tile of C using V_WMMA_F32_16X16X4_F32.
// A layout (16x4, MxK): lanes 0-15 -> rows, VGPR0/1 = K{0,1}; lanes 16-31 = K{2,3}.
// B layout (4x16, KxN): lane = column; VGPR0/1 = K{0,1} (lanes 0-15), K{2,3} (16-31).
// C layout: VGPR i -> row i (lanes 0-15) / row i+8 (lanes 16-31), lane%16 = col.
// W is staged in LDS repacked as [k/4][col][k%4] (zero-padded past K) so each
// lane's two K-values are one aligned b64 load, conflict-free across the wave.

template<int NCOLS, int KMAIN, int KTAIL>
__global__ void __launch_bounds__(256)
k_gemm_wmma(const float* __restrict__ A, const float* __restrict__ W,
            float* __restrict__ C, int M) {
  constexpr int K    = KMAIN + KTAIL;
  constexpr int KPAD = KMAIN + (KTAIL ? 4 : 0);
  constexpr int CT   = NCOLS / 16;   // column tiles
  constexpr int RT   = 8 / CT;       // row tiles per 256-thread (8-wave) block

  __shared__ float wt[(KPAD / 4) * NCOLS * 4];

  // cooperative, coalesced fill of repacked W tile (zero-padded beyond K)
  for (int idx = threadIdx.x; idx < KPAD * NCOLS; idx += 256) {
    const int k   = idx / NCOLS;
    const int col = idx - k * NCOLS;
    const float v = (k < K) ? W[k * NCOLS + col] : 0.0f;
    wt[(k >> 2) * (NCOLS * 4) + col * 4 + (k & 3)] = v;
  }
  __syncthreads();

  const int w    = threadIdx.x >> 5;
  const int lane = threadIdx.x & 31;
  const int lh   = lane >> 4;        // half-wave: selects K pair / +8 row offset
  const int lm   = lane & 15;

  const int tm = blockIdx.x * RT + (w / CT);
  const int tn = w % CT;

  int arow = tm * 16 + lm;
  if (arow >= M) arow = M - 1;       // clamp: garbage rows are never stored
  const float* __restrict__ ap = A + (long long)arow * K + 2 * lh;
  const float* __restrict__ bp = wt + tn * 64 + lm * 4 + 2 * lh;

  v8f c = {};
#pragma unroll
  for (int kk = 0; kk < KMAIN; kk += 4) {
    const v2f a = *(const v2f*)(ap + kk);
    const v2f b = *(const v2f*)(bp + kk * NCOLS);
    c = __builtin_amdgcn_wmma_f32_16x16x4_f32(false, a, false, b,
                                              (short)0, c, false, false);
  }
  if constexpr (KTAIL > 0) {
    const int ka = KMAIN + 2 * lh;
    v2f a;
    a.x = (ka     < K) ? ap[KMAIN]     : 0.0f;   // in-bounds: ap pre-offset by 2*lh
    a.y = (ka + 1 < K) ? ap[KMAIN + 1] : 0.0f;
    const v2f b = *(const v2f*)(bp + KMAIN * NCOLS);
    c = __builtin_amdgcn_wmma_f32_16x16x4_f32(false, a, false, b,
                                              (short)0, c, false, false);
  }

  const int col   = tn * 16 + lm;
  const int rbase = tm * 16 + 8 * lh;
  if (rbase + 7 < M) {               // fast path: uniform, unguarded stores
#pragma unroll
    for (int i = 0; i < 8; ++i)
      C[(long long)(rbase + i) * NCOLS + col] = c[i];
  } else {
#pragma unroll
    for (int i = 0; i < 8; ++i)
      if (rbase + i < M) C[(long long)(rbase + i) * NCOLS + col] = c[i];
  }
}

// ---------------- edge scatter: agg[dst] += xw[src] * dinv[src]*dinv[dst] ---

template<int F, int LF>
__global__ void k_scatter(const float* __restrict__ xw, const int* __restrict__ src,
                          const int* __restrict__ dst, const float* __restrict__ dinv,
                          float* __restrict__ agg, long long nwork) {
  long long idx = (long long)blockIdx.x * blockDim.x + threadIdx.x;
  if (idx >= nwork) return;
  const int       f = (int)(idx & (F - 1));
  const long long e = idx >> LF;
  const int s = src[e];
  const int d = dst[e];
  const float cf = dinv[s] * dinv[d];
  atomicAdd(&agg[(long long)d * F + f], xw[(long long)s * F + f] * cf);
}

// ---------------- self term + bias: agg += xw * dinv^2 + b -----------------

template<int F, int LF>
__global__ void k_self_bias(const float* __restrict__ xw, const float* __restrict__ dinv,
                            const float* __restrict__ b, float* __restrict__ agg,
                            long long nwork) {
  long long idx = (long long)blockIdx.x * blockDim.x + threadIdx.x;
  if (idx >= nwork) return;
  const int f = (int)(idx & (F - 1));
  const int i = (int)(idx >> LF);
  const float di = dinv[i];
  agg[idx] += xw[idx] * di * di + b[f];
}

// ---------------- batch norm: column sums / sumsq --------------------------

template<int F, int LF>
__global__ void k_bn_stats(const float* __restrict__ x, float* __restrict__ sums,
                           float* __restrict__ sq, int n) {
  constexpr int ROWS = 256 / F;
  const int f  = threadIdx.x & (F - 1);
  const int r0 = blockIdx.x * ROWS + (threadIdx.x >> LF);
  float s = 0.0f, q = 0.0f;
  for (int r = r0; r < n; r += (int)gridDim.x * ROWS) {
    const float v = x[(long long)r * F + f];
    s += v;
    q += v * v;
  }
  atomicAdd(&sums[f], s);
  atomicAdd(&sq[f], q);
}

template<int F, int LF>
__global__ void k_bn_apply_elu(const float* __restrict__ x, const float* __restrict__ sums,
                               const float* __restrict__ sq, const float* __restrict__ gamma,
                               const float* __restrict__ beta, float* __restrict__ out,
                               long long nwork, float invN) {
  long long idx = (long long)blockIdx.x * blockDim.x + threadIdx.x;
  if (idx >= nwork) return;
  const int f = (int)(idx & (F - 1));
  const float mu  = sums[f] * invN;
  const float var = sq[f] * invN - mu * mu;
  const float y = gamma[f] * (x[idx] - mu) * rsqrtf(var + GCN_BN_EPS) + beta[f];
  out[idx] = (y > 0.0f) ? y : expm1f(y);
}

// ---------------- pooling --------------------------------------------------

__device__ __forceinline__ unsigned enc_f32(float v) {
  unsigned u = __float_as_uint(v);
  return (u & 0x80000000u) ? ~u : (u | 0x80000000u);
}
__device__ __forceinline__ float dec_f32(unsigned e) {
  return (e & 0x80000000u) ? __uint_as_float(e ^ 0x80000000u) : __uint_as_float(~e);
}

__global__ void k_pool_init(float* __restrict__ psum, unsigned* __restrict__ pmax,
                            float* __restrict__ counts, int ng) {
  int i = blockIdx.x * blockDim.x + threadIdx.x;
  if (i < ng * 32) { psum[i] = 0.0f; pmax[i] = 0x007FFFFFu; /* enc(-inf) */ }
  if (i < ng) counts[i] = 0.0f;
}

__global__ void k_count(const int* __restrict__ batch, float* __restrict__ counts, int n) {
  int i = blockIdx.x * blockDim.x + threadIdx.x;
  if (i < n) atomicAdd(&counts[batch[i]], 1.0f);
}

__global__ void k_pool(const float* __restrict__ h, const int* __restrict__ batch,
                       float* __restrict__ psum, unsigned* __restrict__ pmax,
                       long long nwork) {
  long long idx = (long long)blockIdx.x * blockDim.x + threadIdx.x;
  if (idx >= nwork) return;
  const int i = (int)(idx >> 5);
  const int f = (int)(idx & 31);
  const int b = batch[i];
  const float v = h[idx];
  atomicAdd(&psum[b * 32 + f], v);
  atomicMax(&pmax[b * 32 + f], enc_f32(v));
}

__global__ void k_pool_final(const float* __restrict__ psum, const unsigned* __restrict__ pmax,
                             const float* __restrict__ counts, float* __restrict__ G, int ng) {
  int idx = blockIdx.x * blockDim.x + threadIdx.x;
  if (idx >= ng * 32) return;
  const int g = idx >> 5;
  const int f = idx & 31;
  const float c = counts[g];
  const float mean = psum[idx] / fmaxf(c, 1.0f);
  float mx = dec_f32(pmax[idx]);
  if (!(c > 0.0f)) mx = 0.0f;
  G[g * 64 + f]      = mean;
  G[g * 64 + 32 + f] = mx;
}

// ---------------- MLP head: out = elu(G@Wm1+bm1) @ Wm2 + bm2 ---------------

__global__ void k_mlp(const float* __restrict__ G, const float* __restrict__ Wm1,
                      const float* __restrict__ bm1, const float* __restrict__ Wm2,
                      const float* __restrict__ bm2, float* __restrict__ out) {
  __shared__ float gv[64];
  __shared__ float m1[32];
  const int g = blockIdx.x;
  const int t = threadIdx.x;  // 64 threads
  gv[t] = G[g * 64 + t];
  __syncthreads();
  if (t < 32) {
    float acc = bm1[t];
#pragma unroll
    for (int k = 0; k < 64; ++k) acc += gv[k] * Wm1[k * 32 + t];
    m1[t] = (acc > 0.0f) ? acc : expm1f(acc);
  }
  __syncthreads();
  if (t < 2) {
    float acc = bm2[t];
#pragma unroll
    for (int k = 0; k < 32; ++k) acc += m1[k] * Wm2[k * 2 + t];
    out[g * 2 + t] = acc;
  }
}

// ---------------------------------------------------------------------------

static inline unsigned blocks_for(long long n, int bs) {
  return (unsigned)((n + bs - 1) / bs);
}

extern "C" void kernel_launch(void* const* d_in, const int* in_sizes, int n_in,
                              void* d_out, int out_size, void* d_ws, size_t ws_size,
                              hipStream_t stream) {
  const float* x     = (const float*)d_in[0];
  const int*   ei    = (const int*)d_in[1];
  const int*   batch = (const int*)d_in[3];
  const float* W1    = (const float*)d_in[4];
  const float* b1    = (const float*)d_in[5];
  const float* g1    = (const float*)d_in[6];
  const float* be1   = (const float*)d_in[7];
  const float* W2    = (const float*)d_in[8];
  const float* b2    = (const float*)d_in[9];
  const float* g2    = (const float*)d_in[10];
  const float* be2   = (const float*)d_in[11];
  const float* Wm1   = (const float*)d_in[12];
  const float* bm1   = (const float*)d_in[13];
  const float* Wm2   = (const float*)d_in[14];
  const float* bm2   = (const float*)d_in[15];

  const int NN = in_sizes[3];                 // 100000 nodes
  const long long NE = in_sizes[1] / 2;       // 1.6M edges
  const int* src = ei;
  const int* dst = ei + NE;

  // ---- workspace carve-up (floats) ----
  float* ws = (float*)d_ws;
  size_t off = 0;
  float* bufA = ws + off; off += (size_t)NN * 64;   // xw1 -> h1 -> agg2
  float* bufB = ws + off; off += (size_t)NN * 64;   // agg1 -> xw2 -> h2
  float* dinv = ws + off; off += (size_t)NN;
  float* sums1 = ws + off; off += 64;
  float* sq1   = ws + off; off += 64;
  float* sums2 = ws + off; off += 32;
  float* sq2   = ws + off; off += 32;
  float* psum  = ws + off; off += (size_t)N_GRAPHS_C * 32;
  unsigned* pmax = (unsigned*)(ws + off); off += (size_t)N_GRAPHS_C * 32;
  float* counts = ws + off; off += N_GRAPHS_C;
  float* G      = ws + off; off += (size_t)N_GRAPHS_C * 64;

  const int BS = 256;

  // ---- degree / dinv ----
  k_zero_f<<<blocks_for(NN, BS), BS, 0, stream>>>(dinv, NN);
  k_deg<<<blocks_for(NE, BS), BS, 0, stream>>>(dst, dinv, (int)NE);
  k_dinv<<<blocks_for(NN, BS), BS, 0, stream>>>(dinv, NN);

  // ---- layer 1: xw1 = x @ W1  (M=NN, K=66, N=64) ----
  k_gemm_wmma<64, 64, 2><<<blocks_for(NN, 32), BS, 0, stream>>>(x, W1, bufA, NN);

  // agg1 = scatter + self + bias
  const long long nw64 = (long long)NN * 64;
  k_zero_f<<<blocks_for(nw64, BS), BS, 0, stream>>>(bufB, nw64);
  k_scatter<64, 6><<<blocks_for(NE * 64, BS), BS, 0, stream>>>(bufA, src, dst, dinv, bufB, NE * 64);
  k_self_bias<64, 6><<<blocks_for(nw64, BS), BS, 0, stream>>>(bufA, dinv, b1, bufB, nw64);

  // BN1 + ELU -> h1 in bufA
  k_zero_f<<<1, 192, 0, stream>>>(sums1, 192);  // sums1..sq2 contiguous
  k_bn_stats<64, 6><<<512, BS, 0, stream>>>(bufB, sums1, sq1, NN);
  k_bn_apply_elu<64, 6><<<blocks_for(nw64, BS), BS, 0, stream>>>(
      bufB, sums1, sq1, g1, be1, bufA, nw64, 1.0f / (float)NN);

  // ---- layer 2: xw2 = h1 @ W2  (M=NN, K=64, N=32) ----
  k_gemm_wmma<32, 64, 0><<<blocks_for(NN, 64), BS, 0, stream>>>(bufA, W2, bufB, NN);

  const long long nw32 = (long long)NN * 32;
  k_zero_f<<<blocks_for(nw32, BS), BS, 0, stream>>>(bufA, nw32);  // agg2 in bufA
  k_scatter<32, 5><<<blocks_for(NE * 32, BS), BS, 0, stream>>>(bufB, src, dst, dinv, bufA, NE * 32);
  k_self_bias<32, 5><<<blocks_for(nw32, BS), BS, 0, stream>>>(bufB, dinv, b2, bufA, nw32);

  // BN2 + ELU -> h2 in bufB
  k_bn_stats<32, 5><<<512, BS, 0, stream>>>(bufA, sums2, sq2, NN);
  k_bn_apply_elu<32, 5><<<blocks_for(nw32, BS), BS, 0, stream>>>(
      bufA, sums2, sq2, g2, be2, bufB, nw32, 1.0f / (float)NN);

  // ---- pooling ----
  k_pool_init<<<blocks_for(N_GRAPHS_C * 32, BS), BS, 0, stream>>>(psum, pmax, counts, N_GRAPHS_C);
  k_count<<<blocks_for(NN, BS), BS, 0, stream>>>(batch, counts, NN);
  k_pool<<<blocks_for(nw32, BS), BS, 0, stream>>>(bufB, batch, psum, pmax, nw32);
  k_pool_final<<<blocks_for(N_GRAPHS_C * 32, BS), BS, 0, stream>>>(psum, pmax, counts, G, N_GRAPHS_C);

  // ---- MLP head ----
  k_mlp<<<N_GRAPHS_C, 64, 0, stream>>>(G, Wm1, bm1, Wm2, bm2, (float*)d_out);
}